// BinaryTreeLSTM_29824252903912
// MI455X (gfx1250) — compile-verified
//
#include <hip/hip_runtime.h>
#include <hip/hip_bf16.h>

// CDNA5 / gfx1250 BinaryTreeLSTM, level-wise, fp32 WMMA (V_WMMA_F32_16X16X4_F32).

typedef __attribute__((ext_vector_type(2))) float v2f;
typedef __attribute__((ext_vector_type(8))) float v8f;

#define INPUT_SZ 64
#define HID 128
#define GATES4 512   // 4*HID

__device__ __forceinline__ float sigmoidf_(float v) {
    return 1.0f / (1.0f + __expf(-v));
}

__global__ __launch_bounds__(256)
void tree_lstm_level_kernel(const float* __restrict__ x,
                            const float* __restrict__ W_ih,   // [512][64]
                            const float* __restrict__ W_hh,   // [512][128]
                            const float* __restrict__ b_ih,   // [512]
                            const float* __restrict__ b_hh,   // [512]
                            float* __restrict__ h_out,        // [n][128] == d_out
                            float* __restrict__ c_buf,        // [n][128] in d_ws
                            int start, int count, int is_leaf)
{
    __shared__ float xs[16 * INPUT_SZ];      //  4 KB: x rows for 16 nodes
    __shared__ float hprev[16 * HID];        //  8 KB: assembled h_prev rows
    __shared__ float cprev[16 * HID];        //  8 KB: assembled c_prev rows
    __shared__ float gates[16 * GATES4];     // 32 KB: gate outputs for 16 nodes

    const int tid  = threadIdx.x;
    const int m0   = blockIdx.x * 16;        // first node row of this block's tile
    const int lane = tid & 31;
    const int wave = tid >> 5;               // 0..7
    const int lh   = lane >> 4;              // lane half: 0 or 1 (K pair select)
    const int l    = lane & 15;              // M (for A) / N (for B) index

    // ---- Phase 0: stage per-node operands into LDS (clamped, full block) ----
    {
        // 16 rows x 64 floats = 256 float4 chunks; one per thread
        int m  = tid >> 4;              // 0..15
        int k4 = (tid & 15) << 2;       // 0,4,...,60
        int mm = m0 + m; if (mm >= count) mm = count - 1;
        const float4 v = *(const float4*)(x + (size_t)(start + mm) * INPUT_SZ + k4);
        *(float4*)(xs + m * INPUT_SZ + k4) = v;
    }
    if (!is_leaf) {
        // h_prev[m][j] = j<64 ? h[2*node+1][j] : h[2*node+2][j-64]; same for c_prev.
        // 16 rows x 128 floats = 512 float4 chunks; two per thread.
        #pragma unroll
        for (int rep = 0; rep < 2; ++rep) {
            int idx = tid + rep * 256;         // 0..511
            int m   = idx >> 5;                // 32 chunks per row
            int j4  = (idx & 31) << 2;         // 0..124, never straddles the 64 split
            int mm  = m0 + m; if (mm >= count) mm = count - 1;
            int node  = start + mm;
            int child = (j4 < 64) ? (2 * node + 1) : (2 * node + 2);
            int col   = (j4 < 64) ? j4 : (j4 - 64);
            *(float4*)(hprev + m * HID + j4) =
                *(const float4*)(h_out + (size_t)child * HID + col);
            *(float4*)(cprev + m * HID + j4) =
                *(const float4*)(c_buf + (size_t)child * HID + col);
        }
    }
    __syncthreads();

    // ---- Phase 1: gates = X @ W_ih^T (+ Hprev @ W_hh^T), fp32 WMMA ----
    // Wave `wave` computes gate columns [64*wave, 64*wave+64) as 4 16x16 tiles.
    const int nbase = wave * 64;
    v8f acc[4] = {};

    // X contribution: K = 64 -> 16 k-steps of 4
    #pragma unroll 4
    for (int ks = 0; ks < INPUT_SZ / 4; ++ks) {
        const int koff = ks * 4 + lh * 2;
        // A fragment: lane l holds X[m0+l][koff], X[m0+l][koff+1]
        v2f a = *(const v2f*)(xs + l * INPUT_SZ + koff);
        #pragma unroll
        for (int nt = 0; nt < 4; ++nt) {
            const int row = nbase + nt * 16 + l;     // gate column = W row
            v2f b = *(const v2f*)(W_ih + (size_t)row * INPUT_SZ + koff);
            acc[nt] = __builtin_amdgcn_wmma_f32_16x16x4_f32(
                false, a, false, b, (short)0, acc[nt], false, false);
        }
    }

    // Hprev contribution: K = 128 -> 32 k-steps of 4
    if (!is_leaf) {
        #pragma unroll 4
        for (int ks = 0; ks < HID / 4; ++ks) {
            const int koff = ks * 4 + lh * 2;
            v2f a = *(const v2f*)(hprev + l * HID + koff);
            #pragma unroll
            for (int nt = 0; nt < 4; ++nt) {
                const int row = nbase + nt * 16 + l;
                v2f b = *(const v2f*)(W_hh + (size_t)row * HID + koff);
                acc[nt] = __builtin_amdgcn_wmma_f32_16x16x4_f32(
                    false, a, false, b, (short)0, acc[nt], false, false);
            }
        }
    }

    // D layout: acc[nt][r] = gates[M = r + 8*lh][N = nbase + nt*16 + l]
    #pragma unroll
    for (int nt = 0; nt < 4; ++nt) {
        const int n = nbase + nt * 16 + l;
        #pragma unroll
        for (int r = 0; r < 8; ++r) {
            gates[(r + 8 * lh) * GATES4 + n] = acc[nt][r];
        }
    }
    __syncthreads();

    // ---- Phase 2: LSTM elementwise; 16 nodes x 128 hidden = 2048 elems ----
    #pragma unroll
    for (int rep = 0; rep < 8; ++rep) {
        int idx  = tid + rep * 256;      // 0..2047
        int m    = idx >> 7;             // node row in tile
        int hc   = idx & 127;            // hidden column
        int mm   = m0 + m;
        if (mm < count) {
            int node = start + mm;
            const float* gm = gates + m * GATES4;
            float gi = gm[hc]           + b_ih[hc]           + b_hh[hc];
            float gf = gm[HID + hc]     + b_ih[HID + hc]     + b_hh[HID + hc];
            float gg = gm[2 * HID + hc] + b_ih[2 * HID + hc] + b_hh[2 * HID + hc];
            float go = gm[3 * HID + hc] + b_ih[3 * HID + hc] + b_hh[3 * HID + hc];

            float ii = sigmoidf_(gi);
            float ff = sigmoidf_(gf);
            float oo = sigmoidf_(go);
            float tg = tanhf(gg);
            float cp = is_leaf ? 0.0f : cprev[m * HID + hc];

            float c  = ff * cp + ii * tg;
            float hv = oo * tanhf(c);

            c_buf[(size_t)node * HID + hc] = c;
            h_out[(size_t)node * HID + hc] = hv;
        }
    }
}

extern "C" void kernel_launch(void* const* d_in, const int* in_sizes, int n_in,
                              void* d_out, int out_size, void* d_ws, size_t ws_size,
                              hipStream_t stream) {
    (void)n_in; (void)out_size; (void)ws_size;
    const float* x    = (const float*)d_in[0];   // [n][64]
    const float* W_ih = (const float*)d_in[1];   // [512][64]
    const float* W_hh = (const float*)d_in[2];   // [512][128]
    const float* b_ih = (const float*)d_in[3];   // [512]
    const float* b_hh = (const float*)d_in[4];   // [512]
    float* h_out = (float*)d_out;                // [n][128] (== out_h)
    float* c_buf = (float*)d_ws;                 // [n][128] scratch (needs n*512 bytes)

    const int n = in_sizes[0] / INPUT_SZ;        // 2^L - 1
    int levels = 0;
    while (((1 << levels) - 1) < n) ++levels;    // L = 18 for the reference shapes

    // Leaves -> root; stream ordering provides the level dependency.
    for (int lvl = levels - 1; lvl >= 0; --lvl) {
        const int start = (1 << lvl) - 1;
        const int cnt   = 1 << lvl;
        const int blocks = (cnt + 15) / 16;
        tree_lstm_level_kernel<<<dim3(blocks), dim3(256), 0, stream>>>(
            x, W_ih, W_hh, b_ih, b_hh, h_out, c_buf,
            start, cnt, (lvl == levels - 1) ? 1 : 0);
    }
}